// fast_BEV_55714315764006
// MI455X (gfx1250) — compile-verified
//
#include <hip/hip_runtime.h>
#include <hip/hip_bf16.h>

typedef __attribute__((ext_vector_type(2))) float v2f;
typedef __attribute__((ext_vector_type(8))) float v8f;

#define NCAM   6
#define CIN    256
#define HFEAT  32
#define WFEAT  88
#define HWF    (HFEAT*WFEAT)   // 2816
#define COUT   80
#define NXx    200
#define NYy    200
#define NXY    (NXx*NYy)       // 40000
#define TILE_XY 32
#define BN_EPSf 1e-5f

__global__ __launch_bounds__(256) void fast_bev_fused_kernel(
    const float* __restrict__ feat,   // (2,6,256,32,88)
    const float* __restrict__ l2i,    // (2,6,4,4)
    const float* __restrict__ iam,    // (2,6,4,4)
    const float* __restrict__ lam,    // (2,4,4)
    const float* __restrict__ Wc,     // (80,256)
    const float* __restrict__ bias,   // (80)
    const float* __restrict__ gam,    // (80)
    const float* __restrict__ bet,    // (80)
    const float* __restrict__ mean_,  // (80)
    const float* __restrict__ var_,   // (80)
    float* __restrict__ out)          // (2,80,200,200)
{
    __shared__ int   s_cam[TILE_XY * 4];
    __shared__ int   s_off[TILE_XY * 4];
    __shared__ float s_bev[CIN * TILE_XY];   // 32 KB: [c][xy]

    const int tid    = threadIdx.x;
    const int b      = blockIdx.x / (NXY / TILE_XY);
    const int tile   = blockIdx.x % (NXY / TILE_XY);
    const int xybase = tile * TILE_XY;

    // ---------- Stage 1: project (xy,z) points, pick last-valid camera ----------
    if (tid < TILE_XY * 4) {
        const int xyl = tid >> 2;
        const int z   = tid & 3;
        const int xy  = xybase + xyl;
        const int x   = xy / NYy;
        const int y   = xy % NYy;
        const float px = x * 0.5f - 50.0f;       // voxel grid -> lidar coords
        const float py = y * 0.5f - 50.0f;
        const float pz = z * 1.5f - 4.7f;

        const float* L = lam + b * 16;
        const float dx = px - L[3], dy = py - L[7], dz = pz - L[11];
        // p' = R^T (p - t)
        const float p0 = L[0]*dx + L[4]*dy + L[8] *dz;
        const float p1 = L[1]*dx + L[5]*dy + L[9] *dz;
        const float p2 = L[2]*dx + L[6]*dy + L[10]*dz;

        int best = -1, boff = 0;
        for (int j = 0; j < NCAM; ++j) {
            const float* M = l2i + (size_t)(b*NCAM + j) * 16;
            const float* A = iam + (size_t)(b*NCAM + j) * 16;
            // q = l2i @ [p';1]
            const float q0 = M[0]*p0 + M[1]*p1 + M[2] *p2 + M[3];
            const float q1 = M[4]*p0 + M[5]*p1 + M[6] *p2 + M[7];
            const float q2 = M[8]*p0 + M[9]*p1 + M[10]*p2 + M[11];
            // rows of iam with translation column zeroed
            const float r0 = A[0]*q0 + A[1]*q1 + A[2] *q2;
            const float r1 = A[4]*q0 + A[5]*q1 + A[6] *q2;
            const float Z  = A[8]*q0 + A[9]*q1 + A[10]*q2;
            const float u = r0 / Z + A[3];
            const float v = r1 / Z + A[7];
            const int uf = (int)rintf(u * 0.125f);   // stride = 8
            const int vf = (int)rintf(v * 0.125f);
            if (uf >= 0 && vf >= 0 && uf < WFEAT && vf < HFEAT && Z > 0.0f) {
                best = j;
                boff = vf * WFEAT + uf;
            }
        }
        s_cam[tid] = best;
        s_off[tid] = boff;
    }
    __syncthreads();

    // ---------- Stage 2: gather 256 channels per xy, sum over z -> LDS tile ----------
    {
        const int xyl = tid & (TILE_XY - 1);
        const int cg  = tid >> 5;                  // 8 groups of 32 channels
        const float* basez[4];
        #pragma unroll
        for (int z = 0; z < 4; ++z) {
            const int cam = s_cam[xyl * 4 + z];
            basez[z] = (cam >= 0)
                ? feat + (size_t)((b*NCAM + cam) * CIN) * HWF + s_off[xyl * 4 + z]
                : nullptr;
        }
        for (int cc = 0; cc < 32; ++cc) {
            const int c = cg * 32 + cc;
            float s = 0.0f;
            #pragma unroll
            for (int z = 0; z < 4; ++z)
                if (basez[z]) s += basez[z][(size_t)c * HWF];
            s_bev[c * TILE_XY + xyl] = s;
        }
    }
    __syncthreads();

    // ---------- Stage 3: (80x256)@(256x32) GEMM via v_wmma_f32_16x16x4_f32 ----------
    const int wave = tid >> 5;
    const int lane = tid & 31;
    const int half = lane >> 4;     // 0: lanes 0-15, 1: lanes 16-31
    const int n    = lane & 15;

    // 10 tile jobs = 5 M-tiles (16 out-ch) x 2 N-tiles (16 cols); wave-uniform loop
    for (int j = wave; j < 10; j += 8) {
        const int mtile = j >> 1;
        const int ntile = j & 1;
        const int m     = mtile * 16 + n;          // A-fragment row (lane = M)
        const int ncol  = ntile * 16 + n;          // B/C/D column (lane%16 = N)
        v8f acc = {};
        #pragma unroll 4
        for (int k0 = 0; k0 < CIN; k0 += 4) {
            // A 16x4 f32 layout: VGPR0 K = 2*half, VGPR1 K = 2*half+1
            const v2f a = *(const v2f*)(Wc + (size_t)m * CIN + k0 + half * 2);
            // B 4x16 f32 layout: VGPR0 K = half, VGPR1 K = half+2
            v2f bb;
            bb.x = s_bev[(k0 + half)     * TILE_XY + ncol];
            bb.y = s_bev[(k0 + half + 2) * TILE_XY + ncol];
            acc = __builtin_amdgcn_wmma_f32_16x16x4_f32(
                      false, a, false, bb, (short)0, acc, false, false);
        }
        // C/D layout: VGPR r holds M = r (lanes 0-15) / r+8 (lanes 16-31)
        #pragma unroll
        for (int r = 0; r < 8; ++r) {
            const int o = mtile * 16 + r + half * 8;
            float val = acc[r] + bias[o];
            const float scale = gam[o] * rsqrtf(var_[o] + BN_EPSf);
            val = (val - mean_[o]) * scale + bet[o];
            val = fmaxf(val, 0.0f);
            out[((size_t)(b * COUT + o)) * NXY + xybase + ncol] = val;
        }
    }
}

extern "C" void kernel_launch(void* const* d_in, const int* in_sizes, int n_in,
                              void* d_out, int out_size, void* d_ws, size_t ws_size,
                              hipStream_t stream) {
    (void)in_sizes; (void)n_in; (void)d_ws; (void)ws_size; (void)out_size;
    const float* feat  = (const float*)d_in[0];  // mlvl_feats
    const float* l2i   = (const float*)d_in[1];  // lidar2image
    const float* iam   = (const float*)d_in[2];  // img_aug_matrix
    const float* lam   = (const float*)d_in[3];  // lidar_aug_matrix
    const float* Wc    = (const float*)d_in[4];  // conv_w
    const float* bias  = (const float*)d_in[5];  // conv_b
    const float* gam   = (const float*)d_in[6];  // bn_gamma
    const float* bet   = (const float*)d_in[7];  // bn_beta
    const float* mean_ = (const float*)d_in[8];  // bn_mean
    const float* var_  = (const float*)d_in[9];  // bn_var
    float* out = (float*)d_out;

    const int blocks = 2 * (NXY / TILE_XY);      // 2500
    fast_bev_fused_kernel<<<blocks, 256, 0, stream>>>(
        feat, l2i, iam, lam, Wc, bias, gam, bet, mean_, var_, out);
}